// WarpingLayer_2602750181826
// MI455X (gfx1250) — compile-verified
//
#include <hip/hip_runtime.h>

// Problem constants from the reference
#define Bn 4
#define Cn 64
#define Hn 512
#define Wn 896
#define HWn (Hn * Wn)
#define TILE 224   // 7 wave32s; 896 = 4 * 224 exactly

__global__ __launch_bounds__(TILE) void warp_bilinear_kernel(
    const float* __restrict__ x,     // (B, C, H, W)
    const float* __restrict__ flow,  // (B, 2, H, W)
    float* __restrict__ out)         // (B, C, H, W)
{
    __shared__ float sflow[2 * TILE];

    const int tid  = threadIdx.x;
    const int bid  = blockIdx.x;          // = ((b*H + y)*4 + tile) ; batch-major for L2 locality
    const int tile = bid & 3;
    const int y    = (bid >> 2) & (Hn - 1);
    const int b    = bid >> 11;           // 4 tiles * 512 rows = 2^11 blocks per batch
    const int xp   = tile * TILE + tid;

    // ---- CDNA5 async copy: stage this block's flow (fx, fy) into LDS ----
    // Tracked by ASYNCcnt; completion enforced with s_wait_asynccnt 0.
    const float* gfx = flow + (size_t)b * 2 * HWn + (size_t)y * Wn + xp; // flow[b,0,y,xp]
    const float* gfy = gfx + HWn;                                       // flow[b,1,y,xp]
    unsigned l0 = (unsigned)(size_t)&sflow[tid];         // low 32 bits of generic ptr = LDS byte offset
    unsigned l1 = (unsigned)(size_t)&sflow[TILE + tid];
    asm volatile("global_load_async_to_lds_b32 %0, %1, off"
                 :: "v"(l0), "v"(gfx) : "memory");
    asm volatile("global_load_async_to_lds_b32 %0, %1, off"
                 :: "v"(l1), "v"(gfy) : "memory");
    asm volatile("s_wait_asynccnt 0" ::: "memory");

    const float fx = sflow[tid];
    const float fy = sflow[TILE + tid];

    // ---- per-pixel sample coordinates (exact reference formula order) ----
    const float vx = (float)xp + fx;
    const float vy = (float)y  + fy;
    const float gx = 2.0f * vx / (float)(Wn - 1) - 1.0f;
    const float gy = 2.0f * vy / (float)(Hn - 1) - 1.0f;
    float ix = ((gx + 1.0f) * (float)Wn - 1.0f) * 0.5f;
    float iy = ((gy + 1.0f) * (float)Hn - 1.0f) * 0.5f;
    ix = fminf(fmaxf(ix, 0.0f), (float)(Wn - 1));  // padding_mode='border'
    iy = fminf(fmaxf(iy, 0.0f), (float)(Hn - 1));

    const float x0f = floorf(ix);
    const float y0f = floorf(iy);
    float wx1 = ix - x0f;
    float wy1 = iy - y0f;
    int   x0  = (int)x0f;
    int   y0  = (int)y0f;

    // Border remap: ix==W-1 exactly => x0=W-1, wx1=0. Shift to x0=W-2, wx1=1:
    // mathematically identical (weights are exact 0/1), and guarantees
    // x1=x0+1 and y1=y0+1 are always in-bounds -> taps form a 2x2 block
    // at {i0, i0+1, i0+W, i0+W+1}, loadable as two paired 8-byte loads.
    if (x0 >= Wn - 1) { x0 = Wn - 2; wx1 = 1.0f; }
    if (y0 >= Hn - 1) { y0 = Hn - 2; wy1 = 1.0f; }

    const float wx0 = 1.0f - wx1;
    const float wy0 = 1.0f - wy1;
    const float w00 = wy0 * wx0;
    const float w01 = wy0 * wx1;
    const float w10 = wy1 * wx0;
    const float w11 = wy1 * wx1;

    const int i0 = y0 * Wn + x0;        // <= HW - W - 2
    const int i1 = i0 + Wn;             // <= HW - 2 ; i1+1 in-bounds

    const float* xb = x   + (size_t)b * Cn * HWn;
    float*       ob = out + (size_t)b * Cn * HWn + (size_t)y * Wn + xp;

    // ---- 64-channel gather+blend loop; weights amortized across channels ----
    // Two 8B tap loads per channel (top pair / bottom pair). Nontemporal
    // stores keep the streamed output from evicting the L2-resident x batch
    // slice (117 MB < 192 MB global L2).
#pragma unroll 8
    for (int c = 0; c < Cn; ++c) {
        const float* p = xb + (size_t)c * HWn;
        float2 t0, t1;                       // {t00,t01}, {t10,t11}
        __builtin_memcpy(&t0, p + i0, 8);    // 4B-aligned; unaligned-mode b64 load
        __builtin_memcpy(&t1, p + i1, 8);
        const float v = fmaf(w00, t0.x, fmaf(w01, t0.y,
                        fmaf(w10, t1.x, w11 * t1.y)));
        __builtin_nontemporal_store(v, ob + (size_t)c * HWn);
    }
}

extern "C" void kernel_launch(void* const* d_in, const int* in_sizes, int n_in,
                              void* d_out, int out_size, void* d_ws, size_t ws_size,
                              hipStream_t stream) {
    (void)in_sizes; (void)n_in; (void)out_size; (void)d_ws; (void)ws_size;
    const float* x    = (const float*)d_in[0];
    const float* flow = (const float*)d_in[1];
    float*       out  = (float*)d_out;

    const int blocks = Bn * Hn * (Wn / TILE);  // 8192 blocks, 224 threads each
    warp_bilinear_kernel<<<blocks, TILE, 0, stream>>>(x, flow, out);
}